// SeqCondAttention_31903017074718
// MI455X (gfx1250) — compile-verified
//
#include <hip/hip_runtime.h>
#include <hip/hip_bf16.h>
#include <math.h>

// ---- problem constants (from reference) ----
#define Bc 4
#define Lc 4096
#define Dc 768
#define Kc 12
#define Hc 64
#define BLr (Bc*Lc)          // 16384 rows
#define ZW  780              // MEM + K
#define CW  1548             // merged width: K + 768 + 768
#define DSW 384
#define DEXP 192
#define NCHUNK 32
#define CLEN  128            // Lc / NCHUNK

#define WAVES 4              // waves per block for GEMM kernels

typedef float v2f __attribute__((ext_vector_type(2)));
typedef float v8f __attribute__((ext_vector_type(8)));

__device__ __forceinline__ v8f wmma4(v2f a, v2f b, v8f c) {
  // D = A(16x4 f32) * B(4x16 f32) + C(16x16 f32)   -> v_wmma_f32_16x16x4_f32
  return __builtin_amdgcn_wmma_f32_16x16x4_f32(false, a, false, b, (short)0, c, false, false);
}

__device__ __forceinline__ float softplusf(float x) {
  return (x > 20.f) ? x : log1pf(expf(x));
}

// ------------------------------------------------------------------
// Generic row-major f32 WMMA GEMM: C[M x N] = A[M x K] * B[K x N]
// grid.y * WAVES * 16 == M exactly (M = 16384 always here).
// N tail guarded. K must be a multiple of 4 (all our K are).
// ------------------------------------------------------------------
template<int NT>
__global__ void gemm_wmma(const float* __restrict__ A, const float* __restrict__ B,
                          float* __restrict__ C, int N, int K) {
  const int lane = threadIdx.x & 31;
  const int wv   = threadIdx.x >> 5;
  const int half = lane >> 4;          // 0 or 1
  const int l16  = lane & 15;
  const int mbase = (blockIdx.y * WAVES + wv) * 16;
  const int kk = half * 2;             // this lane-group's K pair base
  const float* Arow = A + (size_t)(mbase + l16) * K;
  v8f acc[NT] = {};
  for (int k0 = 0; k0 < K; k0 += 4) {
    v2f a;
    a.x = Arow[k0 + kk];
    a.y = Arow[k0 + kk + 1];
#pragma unroll
    for (int t = 0; t < NT; ++t) {
      int n = (blockIdx.x * NT + t) * 16 + l16;
      v2f b;
      if (n < N) {
        const float* Bp = B + (size_t)(k0 + kk) * N + n;
        b.x = Bp[0];
        b.y = Bp[N];
      } else { b.x = 0.f; b.y = 0.f; }
      acc[t] = wmma4(a, b, acc[t]);
    }
  }
#pragma unroll
  for (int t = 0; t < NT; ++t) {
    int col = (blockIdx.x * NT + t) * 16 + l16;
    if (col >= N) continue;
#pragma unroll
    for (int r = 0; r < 8; ++r) {
      int row = mbase + r + 8 * half;
      C[(size_t)row * N + col] = acc[t][r];
    }
  }
}

// ------------------------------------------------------------------
// Fused causal conv (CK=4) + decay/score/rotation -> merged (BL x 1548)
// merged = [ p_w (12) | re_k (768) | im_k (768) ]
// ------------------------------------------------------------------
__global__ void merge_kernel(const float* __restrict__ z, const float* __restrict__ ck,
                             const float* __restrict__ sscale, const float* __restrict__ dslope,
                             const float* __restrict__ traw, float* __restrict__ merged) {
  int tid = blockIdx.x * blockDim.x + threadIdx.x;
  if (tid >= BLr * Dc) return;
  int c  = tid % Dc;
  int bl = tid / Dc;
  int l  = bl & (Lc - 1);
  int k  = c >> 6;
  float kv = 0.f, sr = 0.f;
#pragma unroll
  for (int i = 0; i < 4; ++i) {
    int ll = l - 3 + i;
    if (ll >= 0) {
      size_t zr = (size_t)(bl - 3 + i) * ZW;
      kv += z[zr + c]            * ck[i * ZW + c];
      sr += z[zr + 768 + k]      * ck[i * ZW + 768 + k];
    }
  }
  float slope = softplusf(dslope[k]);
  float dist  = (float)((Lc - 1) - l);
  float logp  = fminf(fmaxf(sscale[k] * sr, -20.f), 20.f);
  float pw    = expf(logp - slope * dist);
  float theta = softplusf(traw[c]) + 0.001f;
  float phi   = tanhf(kv) * theta;
  float sp, cp;
  sincosf(phi, &sp, &cp);
  float w = kv * pw;
  size_t base = (size_t)bl * CW;
  merged[base + Kc + c]       = w * cp;
  merged[base + Kc + 768 + c] = w * sp;
  if ((c & 63) == 0) merged[base + k] = pw;
}

// ------------------------------------------------------------------
// 3-phase chunked cumsum along L (in place on merged).
// ------------------------------------------------------------------
__global__ void scan_p1(const float* __restrict__ merged, float* __restrict__ csum) {
  int tid = blockIdx.x * blockDim.x + threadIdx.x;
  if (tid >= Bc * NCHUNK * CW) return;
  int ch    = tid % CW;
  int chunk = (tid / CW) & (NCHUNK - 1);
  int b     = tid / (NCHUNK * CW);
  size_t base = ((size_t)b * Lc + (size_t)chunk * CLEN) * CW + ch;
  float s = 0.f;
  for (int j = 0; j < CLEN; j += 8) {
    size_t i0 = base + (size_t)j * CW;
    float v0 = merged[i0];
    float v1 = merged[i0 + 1 * (size_t)CW];
    float v2 = merged[i0 + 2 * (size_t)CW];
    float v3 = merged[i0 + 3 * (size_t)CW];
    float v4 = merged[i0 + 4 * (size_t)CW];
    float v5 = merged[i0 + 5 * (size_t)CW];
    float v6 = merged[i0 + 6 * (size_t)CW];
    float v7 = merged[i0 + 7 * (size_t)CW];
    s += v0; s += v1; s += v2; s += v3; s += v4; s += v5; s += v6; s += v7;
  }
  csum[tid] = s;
}

__global__ void scan_p2(float* __restrict__ csum) {
  int tid = blockIdx.x * blockDim.x + threadIdx.x;
  if (tid >= Bc * CW) return;
  int ch = tid % CW;
  int b  = tid / CW;
  float run = 0.f;
  for (int chunk = 0; chunk < NCHUNK; ++chunk) {
    size_t i = ((size_t)b * NCHUNK + chunk) * CW + ch;
    float v = csum[i];
    csum[i] = run;      // exclusive prefix of chunk sums
    run += v;
  }
}

__global__ void scan_p3(float* __restrict__ merged, const float* __restrict__ csum) {
  int tid = blockIdx.x * blockDim.x + threadIdx.x;
  if (tid >= Bc * NCHUNK * CW) return;
  int ch    = tid % CW;
  int chunk = (tid / CW) & (NCHUNK - 1);
  int b     = tid / (NCHUNK * CW);
  size_t base = ((size_t)b * Lc + (size_t)chunk * CLEN) * CW + ch;
  float run = csum[tid];
  for (int j = 0; j < CLEN; j += 8) {
    size_t i0 = base + (size_t)j * CW;
    float v0 = merged[i0];
    float v1 = merged[i0 + 1 * (size_t)CW];
    float v2 = merged[i0 + 2 * (size_t)CW];
    float v3 = merged[i0 + 3 * (size_t)CW];
    float v4 = merged[i0 + 4 * (size_t)CW];
    float v5 = merged[i0 + 5 * (size_t)CW];
    float v6 = merged[i0 + 6 * (size_t)CW];
    float v7 = merged[i0 + 7 * (size_t)CW];
    run += v0; merged[i0]                   = run;
    run += v1; merged[i0 + 1 * (size_t)CW]  = run;
    run += v2; merged[i0 + 2 * (size_t)CW]  = run;
    run += v3; merged[i0 + 3 * (size_t)CW]  = run;
    run += v4; merged[i0 + 4 * (size_t)CW]  = run;
    run += v5; merged[i0 + 5 * (size_t)CW]  = run;
    run += v6; merged[i0 + 6 * (size_t)CW]  = run;
    run += v7; merged[i0 + 7 * (size_t)CW]  = run;
  }
}

// ------------------------------------------------------------------
// Normalize by cumulative denominator + contract with q (complex rotation)
// ore/oim : (BL x 768), row layout k*64 + h, scaled by norm_scale.
// ------------------------------------------------------------------
__global__ void outreim_kernel(const float* __restrict__ cs, const float* __restrict__ q,
                               const float* __restrict__ nscale,
                               float* __restrict__ ore, float* __restrict__ oim) {
  int tid = blockIdx.x * blockDim.x + threadIdx.x;
  if (tid >= BLr * Dc) return;
  int c  = tid % Dc;
  int bl = tid / Dc;
  int k  = c >> 6, h = c & 63, kq = k >> 1;
  size_t cb = (size_t)bl * CW;
  float den = fmaxf(cs[cb + k], 1e-4f);
  float inv = 1.f / den;
  float sre = cs[cb + Kc + c] * inv;
  float sim = cs[cb + Kc + 768 + c] * inv;
  size_t qb = (size_t)bl * Dc + (size_t)kq * 128 + (size_t)h * 2;
  float qre = q[qb], qim = q[qb + 1];
  float ns = nscale[c];
  ore[tid] = (sre * qre + sim * qim) * ns;
  oim[tid] = (sim * qre - sre * qim) * ns;
}

// ------------------------------------------------------------------
// Batched per-head GEMM: yspec[bl, k*384+n] =
//   sigmoid(graw[bl,k]+bg[k]) * ( ore_k @ W_re[k] + oim_k @ W_im[k] )
// Kdim = 64, N = 384, grid.z = 12.
// ------------------------------------------------------------------
__global__ void gemm_spec(const float* __restrict__ ORE, const float* __restrict__ OIM,
                          const float* __restrict__ Wre, const float* __restrict__ Wim,
                          const float* __restrict__ graw, const float* __restrict__ bgate,
                          float* __restrict__ ysp) {
  const int lane = threadIdx.x & 31;
  const int wv   = threadIdx.x >> 5;
  const int half = lane >> 4;
  const int l16  = lane & 15;
  const int k    = blockIdx.z;
  const int mbase = (blockIdx.y * WAVES + wv) * 16;
  const int kk = half * 2;
  const float* A1 = ORE + (size_t)(mbase + l16) * Dc + (size_t)k * Hc;
  const float* A2 = OIM + (size_t)(mbase + l16) * Dc + (size_t)k * Hc;
  const float* B1 = Wre + (size_t)k * Hc * DSW;
  const float* B2 = Wim + (size_t)k * Hc * DSW;
  v8f acc[4] = {};
  for (int k0 = 0; k0 < Hc; k0 += 4) {
    v2f a1; a1.x = A1[k0 + kk]; a1.y = A1[k0 + kk + 1];
    v2f a2; a2.x = A2[k0 + kk]; a2.y = A2[k0 + kk + 1];
#pragma unroll
    for (int t = 0; t < 4; ++t) {
      int n = (blockIdx.x * 4 + t) * 16 + l16;
      const float* b1 = B1 + (size_t)(k0 + kk) * DSW + n;
      const float* b2 = B2 + (size_t)(k0 + kk) * DSW + n;
      v2f bb;
      bb.x = b1[0]; bb.y = b1[DSW];
      acc[t] = wmma4(a1, bb, acc[t]);
      bb.x = b2[0]; bb.y = b2[DSW];
      acc[t] = wmma4(a2, bb, acc[t]);
    }
  }
  float bgk = bgate[k];
#pragma unroll
  for (int r = 0; r < 8; ++r) {
    int row = mbase + r + 8 * half;
    float g = graw[(size_t)row * Kc + k] + bgk;
    float gate = 1.f / (1.f + expf(-g));
#pragma unroll
    for (int t = 0; t < 4; ++t) {
      int col = (blockIdx.x * 4 + t) * 16 + l16;
      ysp[(size_t)row * (Kc * DSW) + (size_t)k * DSW + col] = acc[t][r] * gate;
    }
  }
}

// ------------------------------------------------------------------
// Highway + SiLU combine: ya (BL x 2304)
// ------------------------------------------------------------------
__global__ void combine_kernel(const float* __restrict__ ysp, const float* __restrict__ u,
                               const float* __restrict__ hsc, float* __restrict__ ya) {
  int tid = blockIdx.x * blockDim.x + threadIdx.x;
  if (tid >= BLr * (Kc * DEXP)) return;
  int c  = tid % (Kc * DEXP);
  int bl = tid / (Kc * DEXP);
  int k = c / DEXP, j = c % DEXP;
  size_t base = (size_t)bl * (Kc * DSW) + (size_t)k * DSW;
  float hs = hsc[k];
  float yv = ysp[base + j]        + u[base + j]        * hs;
  float yg = ysp[base + DEXP + j] + u[base + DEXP + j] * hs;
  float sg = 1.f / (1.f + expf(-yg));
  ya[tid] = yv * yg * sg;
}

// ------------------------------------------------------------------
extern "C" void kernel_launch(void* const* d_in, const int* in_sizes, int n_in,
                              void* d_out, int out_size, void* d_ws, size_t ws_size,
                              hipStream_t stream) {
  const float* x        = (const float*)d_in[0];
  const float* W_mem    = (const float*)d_in[1];
  const float* conv_k   = (const float*)d_in[2];
  const float* W_q      = (const float*)d_in[3];
  const float* traw     = (const float*)d_in[4];
  const float* dslope   = (const float*)d_in[5];
  const float* sscale   = (const float*)d_in[6];
  const float* W_re     = (const float*)d_in[7];
  const float* W_im     = (const float*)d_in[8];
  const float* nscale   = (const float*)d_in[9];
  const float* Wg       = (const float*)d_in[10];
  const float* bg       = (const float*)d_in[11];
  const float* W_down   = (const float*)d_in[12];
  const float* W_up     = (const float*)d_in[13];
  const float* hsc      = (const float*)d_in[14];
  const float* W_out    = (const float*)d_in[15];
  float* out = (float*)d_out;

  float* ws = (float*)d_ws;
  size_t off = 0;
  auto alloc = [&](size_t n) { float* p = ws + off; off += (n + 63) & ~(size_t)63; return p; };
  float* q      = alloc((size_t)BLr * Dc);          // 50 MB
  float* z      = alloc((size_t)BLr * ZW);          // 51 MB
  float* graw   = alloc((size_t)BLr * Kc);          //  1 MB
  float* t      = alloc((size_t)BLr * 192);         // 13 MB
  float* u      = alloc((size_t)BLr * Kc * DSW);    // 302 MB
  float* merged = alloc((size_t)BLr * CW);          // 101 MB
  float* csum   = alloc((size_t)Bc * NCHUNK * CW);  //  1 MB
  float* ore    = alloc((size_t)BLr * Dc);          // 50 MB
  float* oim    = alloc((size_t)BLr * Dc);          // 50 MB
  float* ysp    = alloc((size_t)BLr * Kc * DSW);    // 302 MB
  float* ya     = alloc((size_t)BLr * Kc * DEXP);   // 151 MB

  const dim3 gblk(32 * WAVES);
  const int MY = BLr / (16 * WAVES);                // 256

  // Dense GEMMs on the f32 WMMA path
  gemm_wmma<4><<<dim3((ZW + 63) / 64, MY), gblk, 0, stream>>>(x, W_mem, z, ZW, Dc);
  gemm_wmma<4><<<dim3(Dc / 64, MY), gblk, 0, stream>>>(x, W_q, q, Dc, Dc);
  gemm_wmma<1><<<dim3(1, MY), gblk, 0, stream>>>(x, Wg, graw, Kc, Dc);
  gemm_wmma<4><<<dim3(192 / 64, MY), gblk, 0, stream>>>(x, W_down, t, 192, Dc);
  gemm_wmma<4><<<dim3((Kc * DSW) / 64, MY), gblk, 0, stream>>>(t, W_up, u, Kc * DSW, 192);

  // Conv + rotation -> merged
  {
    int n = BLr * Dc;
    merge_kernel<<<(n + 255) / 256, 256, 0, stream>>>(z, conv_k, sscale, dslope, traw, merged);
  }

  // Chunked cumsum over L (in place)
  {
    int n1 = Bc * NCHUNK * CW;
    scan_p1<<<(n1 + 255) / 256, 256, 0, stream>>>(merged, csum);
    int n2 = Bc * CW;
    scan_p2<<<(n2 + 255) / 256, 256, 0, stream>>>(csum);
    scan_p3<<<(n1 + 255) / 256, 256, 0, stream>>>(merged, csum);
  }

  // Normalize + contract with q
  {
    int n = BLr * Dc;
    outreim_kernel<<<(n + 255) / 256, 256, 0, stream>>>(merged, q, nscale, ore, oim);
  }

  // Per-head spectral GEMM with gate epilogue
  gemm_spec<<<dim3(DSW / 64, MY, Kc), gblk, 0, stream>>>(ore, oim, W_re, W_im, graw, bg, ysp);

  // Highway + SiLU
  {
    int n = BLr * Kc * DEXP;
    combine_kernel<<<(n + 255) / 256, 256, 0, stream>>>(ysp, u, hsc, ya);
  }

  // Final projection
  gemm_wmma<4><<<dim3(Dc / 64, MY), gblk, 0, stream>>>(ya, W_out, out, Dc, Kc * DEXP);
}